// SSM_72730976190810
// MI455X (gfx1250) — compile-verified
//
#include <hip/hip_runtime.h>
#include <math.h>

#define D_INNER 1024
#define STATE   16
#define DT_RANK 64
#define NPROJ   (DT_RANK + 2*STATE)   // 96
#define SEQ_L   1024
#define BATCH   2
#define M_ROWS  (BATCH*SEQ_L)         // 2048
#define LOG_EPS (-7.0f)
#define CLAMP_V (20.0f)
#define LTILE   8                     // timesteps batched per load clause

typedef float v2f __attribute__((ext_vector_type(2)));
typedef float v8f __attribute__((ext_vector_type(8)));

// -------------------------------------------------------------------------
// GEMM1: proj[M=2048, 96] = x[M,1024] @ W_xproj[1024,96]   (fp32 WMMA)
// One wave per 16x16 output tile; K loop steps 4 (V_WMMA_F32_16X16X4_F32).
// A layout (16x4 f32): lane m=lane&15 holds K = 2*(lane>>4) + {0,1}.
// B layout (4x16 f32): lane n=lane&15 holds K = 2*(lane>>4) + {0,1}.
// C/D layout: VGPR r -> M = r + 8*(lane>>4), N = lane&15.
// -------------------------------------------------------------------------
__global__ __launch_bounds__(32)
void ssm_proj_gemm(const float* __restrict__ x,
                   const float* __restrict__ W,
                   float* __restrict__ proj)
{
    const int lane = threadIdx.x;
    const int mlo  = lane & 15;
    const int hi   = lane >> 4;          // 0 or 1
    const int mt   = blockIdx.x;         // 0..127
    const int nt   = blockIdx.y;         // 0..5
    const int row  = mt*16 + mlo;
    const int ncol = nt*16 + mlo;

    const float* arow = x + (size_t)row * D_INNER + 2*hi;       // K pair base
    const float* bcol = W + (size_t)(2*hi) * NPROJ + ncol;

    v8f c = {};
    #pragma unroll 4
    for (int k = 0; k < D_INNER; k += 4) {
        v2f a; a.x = arow[k];                  a.y = arow[k + 1];
        v2f b; b.x = bcol[(size_t)k * NPROJ];  b.y = bcol[(size_t)(k + 1) * NPROJ];
        c = __builtin_amdgcn_wmma_f32_16x16x4_f32(false, a, false, b,
                                                  (short)0, c, false, false);
    }
    #pragma unroll
    for (int r = 0; r < 8; ++r) {
        const int m = mt*16 + r + 8*hi;
        proj[(size_t)m * NPROJ + ncol] = c[r];
    }
}

// -------------------------------------------------------------------------
// GEMM2: delta[M=2048,1024] = softplus(proj[:, :64] @ W_dt[64,1024] + b_dt)
// -------------------------------------------------------------------------
__global__ __launch_bounds__(32)
void ssm_delta_gemm(const float* __restrict__ proj,
                    const float* __restrict__ W_dt,
                    const float* __restrict__ b_dt,
                    float* __restrict__ delta)
{
    const int lane = threadIdx.x;
    const int mlo  = lane & 15;
    const int hi   = lane >> 4;
    const int mt   = blockIdx.x;         // 0..127
    const int nt   = blockIdx.y;         // 0..63
    const int row  = mt*16 + mlo;
    const int ncol = nt*16 + mlo;

    const float* arow = proj + (size_t)row * NPROJ + 2*hi;      // dt_inter cols 0..63
    const float* bcol = W_dt + (size_t)(2*hi) * D_INNER + ncol;

    v8f c = {};
    #pragma unroll
    for (int k = 0; k < DT_RANK; k += 4) {
        v2f a; a.x = arow[k];                    a.y = arow[k + 1];
        v2f b; b.x = bcol[(size_t)k * D_INNER];  b.y = bcol[(size_t)(k + 1) * D_INNER];
        c = __builtin_amdgcn_wmma_f32_16x16x4_f32(false, a, false, b,
                                                  (short)0, c, false, false);
    }
    const float bias = b_dt[ncol];
    #pragma unroll
    for (int r = 0; r < 8; ++r) {
        const int m = mt*16 + r + 8*hi;
        const float z  = c[r] + bias;
        const float sp = (z > 20.f) ? z : log1pf(__expf(z));   // stable softplus
        delta[(size_t)m * D_INNER + ncol] = sp;
    }
}

// -------------------------------------------------------------------------
// Selective scan: 16 lanes per (b,d) channel, one lane per state n.
// 2 channels/wave -> 1024 waves. Software-pipelined: each tile of LTILE
// timesteps is loaded as one clause (one wait), next tile is prefetched
// (global_prefetch_b8) while the current tile computes from registers.
//  lr = cumsum clip(delta*A, -7, 20)            (log_R)
//  s  = cumsum(delta*B*x * exp(min(-lr,20)))
//  y  = sum_n C_n * exp(min(lr,20)) * s_n  + x*D
// -------------------------------------------------------------------------
__global__ __launch_bounds__(256)
void ssm_scan(const float* __restrict__ x,
              const float* __restrict__ proj,
              const float* __restrict__ delta,
              const float* __restrict__ A_log,
              const float* __restrict__ Dvec,
              float* __restrict__ y)
{
    const int tid = blockIdx.x * blockDim.x + threadIdx.x;
    const int ch  = tid >> 4;            // 0..2047
    const int n   = tid & 15;
    const int b   = ch >> 10;
    const int d   = ch & (D_INNER - 1);

    const float A_dn = -__expf(A_log[d * STATE + n]);
    const float Dd   = Dvec[d];

    const float* xp  = x     + (size_t)b * SEQ_L * D_INNER + d;
    const float* dp  = delta + (size_t)b * SEQ_L * D_INNER + d;
    const float* pbc = proj  + (size_t)b * SEQ_L * NPROJ + DT_RANK + n; // B; +STATE -> C
    float*       yp  = y     + (size_t)b * SEQ_L * D_INNER + d;

    float lr = 0.f, s = 0.f;
    for (int lt = 0; lt < SEQ_L; lt += LTILE) {
        // Prefetch next tile into cache while this tile computes.
        if (lt + LTILE < SEQ_L) {
            __builtin_prefetch(&dp [(size_t)(lt + LTILE) * D_INNER], 0, 3);
            __builtin_prefetch(&xp [(size_t)(lt + LTILE) * D_INNER], 0, 3);
            __builtin_prefetch(&pbc[(size_t)(lt + LTILE) * NPROJ  ], 0, 3);
        }

        // One batched load clause for the whole tile.
        float dv[LTILE], xv[LTILE], Bn[LTILE], Cn[LTILE];
        #pragma unroll
        for (int j = 0; j < LTILE; ++j) {
            const size_t l = (size_t)(lt + j);
            dv[j] = dp [l * D_INNER];
            xv[j] = xp [l * D_INNER];
            Bn[j] = pbc[l * NPROJ];
            Cn[j] = pbc[l * NPROJ + STATE];
        }

        // Register-resident recurrence steps.
        #pragma unroll
        for (int j = 0; j < LTILE; ++j) {
            lr += fminf(fmaxf(dv[j] * A_dn, LOG_EPS), CLAMP_V);
            const float enR = __expf(fminf(-lr, CLAMP_V));
            const float R   = __expf(fminf( lr, CLAMP_V));
            s = fmaf(dv[j] * Bn[j] * xv[j], enR, s);

            float contrib = Cn[j] * R * s;
            contrib += __shfl_xor(contrib, 8, 16);
            contrib += __shfl_xor(contrib, 4, 16);
            contrib += __shfl_xor(contrib, 2, 16);
            contrib += __shfl_xor(contrib, 1, 16);

            if (n == 0) yp[(size_t)(lt + j) * D_INNER] = fmaf(xv[j], Dd, contrib);
        }
    }
}

// -------------------------------------------------------------------------
extern "C" void kernel_launch(void* const* d_in, const int* in_sizes, int n_in,
                              void* d_out, int out_size, void* d_ws, size_t ws_size,
                              hipStream_t stream)
{
    (void)in_sizes; (void)n_in; (void)out_size; (void)ws_size;
    const float* x     = (const float*)d_in[0];
    const float* Wxp   = (const float*)d_in[1];
    const float* Wdt   = (const float*)d_in[2];
    const float* bdt   = (const float*)d_in[3];
    const float* A_log = (const float*)d_in[4];
    const float* Dv    = (const float*)d_in[5];
    float* y = (float*)d_out;

    float* proj  = (float*)d_ws;                       // 2048*96 f32
    float* delta = proj + (size_t)M_ROWS * NPROJ;      // 2048*1024 f32

    ssm_proj_gemm <<<dim3(M_ROWS/16, NPROJ/16),   32, 0, stream>>>(x, Wxp, proj);
    ssm_delta_gemm<<<dim3(M_ROWS/16, D_INNER/16), 32, 0, stream>>>(proj, Wdt, bdt, delta);
    ssm_scan      <<<dim3((M_ROWS*STATE)/256),   256, 0, stream>>>(x, proj, delta, A_log, Dv, y);
}